// PermutationInvariantTrainingLoss_17386027614818
// MI455X (gfx1250) — compile-verified
//
#include <hip/hip_runtime.h>
#include <hip/hip_bf16.h>

// ---------------------------------------------------------------------------
// Permutation-Invariant Training loss for MI455X (gfx1250).
//
// Roofline: pred+target+mask ~= 75 MB streamed once -> ~3.2 us floor at
// 23.3 TB/s; MAC work is ~0.27 GFLOP. Pure bandwidth-bound, so:
//   - single coalesced f32 pass over pred/target (no precision downgrade:
//     f16 WMMA would not reduce HBM traffic, inputs are f32 in memory)
//   - T-contraction via V_WMMA_F32_16X16X4_F32, packing 4 independent
//     time-groups block-diagonally in the 16x16 tile (16 t-steps / WMMA pair)
//   - logs computed as bare v_log_f32 (log2, via __builtin_amdgcn_logf, no
//     ln2/NaN fixup code) and the single ln2 factor folded into the final
//     cost-matrix scale, keeping per-byte VALU cost far below what is needed
//     to saturate HBM.
// ---------------------------------------------------------------------------

typedef float v2f __attribute__((ext_vector_type(2)));
typedef float v8f __attribute__((ext_vector_type(8)));

#define PIT_T      16384
#define PIT_S      4
#define PIT_NSLICE 8      // workgroups per batch
#define PIT_NWAVE  8      // waves per workgroup (256 threads, wave32)
#define PIT_EPS    1e-7f
#define PIT_LN2    0.69314718055994530942f

// itertools.permutations(range(4)) order (lexicographic)
__device__ __constant__ int PIT_PERMS[24][4] = {
  {0,1,2,3},{0,1,3,2},{0,2,1,3},{0,2,3,1},{0,3,1,2},{0,3,2,1},
  {1,0,2,3},{1,0,3,2},{1,2,0,3},{1,2,3,0},{1,3,0,2},{1,3,2,0},
  {2,0,1,3},{2,0,3,1},{2,1,0,3},{2,1,3,0},{2,3,0,1},{2,3,1,0},
  {3,0,1,2},{3,0,2,1},{3,1,0,2},{3,1,2,0},{3,2,0,1},{3,2,1,0}
};

// ---------------------------------------------------------------------------
// Kernel 1: streaming BCE cost-matrix accumulation with f32 WMMA.
//
// Time-step assignment tau(g,k) = t0 + 4k + g (g = row/col group 0..3,
// k = WMMA contraction index 0..3).  With the ISA VGPR layouts
// (A: v0 holds K=0|2 across lane halves, v1 holds K=1|3; B mirrors it),
// a lane (=16c+L) needs elements at flat offset
//     4*t0 + (32c + L)          (vgpr .x)
//     4*t0 + (32c + L) + 16     (vgpr .y)
// for speaker s = L%4 -> fully coalesced streaming of the [T,4] planes.
// Valid data lands in the four diagonal 4x4 blocks of the 16x16 accumulator.
// The accumulator holds base-2 log sums; ln2 is applied in finalize.
// ---------------------------------------------------------------------------
__global__ __launch_bounds__(256)
void pit_gemm_kernel(const float* __restrict__ pred,
                     const float* __restrict__ target,
                     const float* __restrict__ mask,
                     float* __restrict__ partials)
{
    const int b     = blockIdx.y;
    const int slice = blockIdx.x;
    const int tid   = threadIdx.x;
    const int w     = tid >> 5;   // wave id 0..7
    const int lane  = tid & 31;

    const float* __restrict__ predb = pred   + (size_t)b * (PIT_T * PIT_S);
    const float* __restrict__ targb = target + (size_t)b * (PIT_T * PIT_S);
    const float* __restrict__ maskb = mask   + (size_t)b * PIT_T;

    const int laneoff = (lane & 15) + ((lane >> 4) << 5);  // 32c + L
    const int moff    = laneoff >> 2;                      // time index within chunk

    v8f acc = {0.f, 0.f, 0.f, 0.f, 0.f, 0.f, 0.f, 0.f};

    const int chunks_per_wave = PIT_T / 16 / (PIT_NSLICE * PIT_NWAVE); // 16

#pragma unroll 2
    for (int k = 0; k < chunks_per_wave; ++k) {
        const int chunk = (k * PIT_NSLICE + slice) * PIT_NWAVE + w; // 0..1023
        const int t0    = chunk << 4;                               // 16 t per chunk
        const int off   = (t0 << 2) + laneoff;

        float p0 = predb[off];
        float p1 = predb[off + 16];
        float g0 = targb[off];
        float g1 = targb[off + 16];
        float m0 = maskb[t0 + moff];
        float m1 = maskb[t0 + moff + 4];

        p0 = fminf(fmaxf(p0, PIT_EPS), 1.0f - PIT_EPS);
        p1 = fminf(fmaxf(p1, PIT_EPS), 1.0f - PIT_EPS);

        // Bare v_log_f32 (base-2 log, no fixup): arguments are clamped to
        // [EPS, 1-EPS], so strictly positive, no denormals/inf/NaN.
        // ln2 conversion is deferred to the finalize scale (exact: the cost
        // matrix is linear in the logs).
        v2f a1; a1.x = __builtin_amdgcn_logf(p0);         // log2 p
        v2f a2; a2.x = __builtin_amdgcn_logf(1.0f - p0);  // log2 (1-p)
        a1.y = __builtin_amdgcn_logf(p1);
        a2.y = __builtin_amdgcn_logf(1.0f - p1);
        v2f b1; b1.x = m0 * g0;           b1.y = m1 * g1;            // m*t
        v2f b2; b2.x = m0 * (1.0f - g0);  b2.y = m1 * (1.0f - g1);   // m*(1-t)

        // acc += A1*B1 + A2*B2  (both terms share one f32 accumulator)
        acc = __builtin_amdgcn_wmma_f32_16x16x4_f32(
                  false, a1, false, b1, (short)0, acc, false, false);
        acc = __builtin_amdgcn_wmma_f32_16x16x4_f32(
                  false, a2, false, b2, (short)0, acc, false, false);
    }

    // Cross-wave reduction of the 4 diagonal blocks via LDS.
    // C/D layout: vgpr r <- row m=r (lanes 0-15, n=lane) and m=r+8 (lanes 16-31,
    // n=lane-16).  Diagonal block (g,g)[i,j] lives at:
    //   g=0: v[i]   lane j      g=1: v[4+i] lane 4+j
    //   g=2: v[i]   lane 24+j   g=3: v[4+i] lane 28+j
    __shared__ float red[PIT_NWAVE][8][32];
#pragma unroll
    for (int r = 0; r < 8; ++r) red[w][r][lane] = acc[r];
    __syncthreads();

    if (tid < 16) {
        const int i = tid >> 2;
        const int j = tid & 3;
        float v = 0.f;
#pragma unroll
        for (int ww = 0; ww < PIT_NWAVE; ++ww) {
            v += red[ww][i][j];
            v += red[ww][4 + i][4 + j];
            v += red[ww][i][24 + j];
            v += red[ww][4 + i][28 + j];
        }
        partials[((size_t)b * PIT_NSLICE + slice) * 16 + tid] = v;
    }
}

// ---------------------------------------------------------------------------
// Kernel 2: per-batch finalize — mask denominator, fixed-order slice sum,
// ln2 * sign * 1/(S*denom) scale, 24-permutation min/argmin (first-min ties
// like jnp.argmin), best perm out.
// ---------------------------------------------------------------------------
__global__ __launch_bounds__(256)
void pit_finalize_kernel(const float* __restrict__ mask,
                         const float* __restrict__ partials,
                         float* __restrict__ minloss,
                         float* __restrict__ out)
{
    const int b   = blockIdx.x;
    const int tid = threadIdx.x;

    __shared__ float sred[256];
    __shared__ float C[16];

    const float* __restrict__ maskb = mask + (size_t)b * PIT_T;
    float s = 0.f;
    for (int t = tid; t < PIT_T; t += 256) s += maskb[t];
    sred[tid] = s;
    __syncthreads();
    for (int st = 128; st > 0; st >>= 1) {
        if (tid < st) sred[tid] += sred[tid + st];
        __syncthreads();
    }

    if (tid < 16) {
        float v = 0.f;
        for (int sl = 0; sl < PIT_NSLICE; ++sl)
            v += partials[((size_t)b * PIT_NSLICE + sl) * 16 + tid];
        C[tid] = v;
    }
    __syncthreads();

    if (tid == 0) {
        const float denom = sred[0];
        // bce sign, /(S*denom), and the deferred log2->ln conversion.
        const float scale = -PIT_LN2 / ((float)PIT_S * denom);
        float Cm[16];
#pragma unroll
        for (int x = 0; x < 16; ++x) Cm[x] = C[x] * scale;

        float best  = 3.402823466e38f;
        int   bestp = 0;
        for (int p = 0; p < 24; ++p) {
            const float pl = Cm[0 * 4 + PIT_PERMS[p][0]]
                           + Cm[1 * 4 + PIT_PERMS[p][1]]
                           + Cm[2 * 4 + PIT_PERMS[p][2]]
                           + Cm[3 * 4 + PIT_PERMS[p][3]];
            if (pl < best) { best = pl; bestp = p; }
        }
        minloss[b] = best;
#pragma unroll
        for (int i = 0; i < 4; ++i)
            out[1 + b * 4 + i] = (float)PIT_PERMS[bestp][i];
    }
}

// ---------------------------------------------------------------------------
// Kernel 3: deterministic mean of per-batch min losses -> out[0].
// ---------------------------------------------------------------------------
__global__ __launch_bounds__(128)
void pit_mean_kernel(const float* __restrict__ minloss,
                     float* __restrict__ out, int B)
{
    __shared__ float sred[128];
    const int tid = threadIdx.x;
    float s = (tid < B) ? minloss[tid] : 0.f;
    sred[tid] = s;
    __syncthreads();
    for (int st = 64; st > 0; st >>= 1) {
        if (tid < st) sred[tid] += sred[tid + st];
        __syncthreads();
    }
    if (tid == 0) out[0] = sred[0] / (float)B;
}

// ---------------------------------------------------------------------------
extern "C" void kernel_launch(void* const* d_in, const int* in_sizes, int n_in,
                              void* d_out, int out_size, void* d_ws, size_t ws_size,
                              hipStream_t stream) {
    const float* pred   = (const float*)d_in[0];
    const float* target = (const float*)d_in[1];
    const float* mask   = (const float*)d_in[2];
    float* out = (float*)d_out;

    int B = in_sizes[2] / PIT_T;   // mask is [B, T]
    if (B <= 0) B = 128;

    // Workspace: partials [B][NSLICE][16] floats, then minloss [B] floats.
    float* partials = (float*)d_ws;
    float* minloss  = partials + (size_t)B * PIT_NSLICE * 16;

    dim3 g1(PIT_NSLICE, B);
    pit_gemm_kernel<<<g1, 256, 0, stream>>>(pred, target, mask, partials);
    pit_finalize_kernel<<<B, 256, 0, stream>>>(mask, partials, minloss, out);
    pit_mean_kernel<<<1, 128, 0, stream>>>(minloss, out, B);
}